// TokenDropper_74741020885694
// MI455X (gfx1250) — compile-verified
//
#include <hip/hip_runtime.h>
#include <hip/hip_bf16.h>

typedef float v2f __attribute__((ext_vector_type(2)));
typedef float v8f __attribute__((ext_vector_type(8)));

#define LN_EPS 1e-5f

// ---------------------------------------------------------------------------
// Kernel 0: gw[d] = gamma[d]*w[d]; S[0] = sum(gamma*w); S[1] = sum(beta*w)+b
// ---------------------------------------------------------------------------
__global__ __launch_bounds__(256)
void prep_kernel(const float* __restrict__ gamma, const float* __restrict__ beta,
                 const float* __restrict__ w, const float* __restrict__ bscal,
                 float* __restrict__ gw, float* __restrict__ S, int D) {
    __shared__ float r1[256];
    __shared__ float r2[256];
    float s1 = 0.f, s2 = 0.f;
    for (int i = threadIdx.x; i < D; i += blockDim.x) {
        float g = gamma[i], ww = w[i];
        gw[i] = g * ww;
        s1 += g * ww;
        s2 += beta[i] * ww;
    }
    r1[threadIdx.x] = s1;
    r2[threadIdx.x] = s2;
    __syncthreads();
    for (int off = 128; off > 0; off >>= 1) {
        if ((int)threadIdx.x < off) {
            r1[threadIdx.x] += r1[threadIdx.x + off];
            r2[threadIdx.x] += r2[threadIdx.x + off];
        }
        __syncthreads();
    }
    if (threadIdx.x == 0) {
        S[0] = r1[0];
        S[1] = r2[0] + bscal[0];
    }
}

// ---------------------------------------------------------------------------
// Kernel 1: fused LayerNorm + scorer. One wave handles 16 tokens.
// Uses V_WMMA_F32_16X16X4_F32 for the GEMV part:
//   A (16x4 fp32): lane%16 = token row M; lanes<16 -> K={0,1}, lanes>=16 -> K={2,3}
//   B (4x16 fp32): gw chunk broadcast across all N columns (value depends only
//                  on K), so every column of D holds the same dot product.
// Per-lane accumulation of sum / sumsq; lanes l and l+16 jointly cover a row.
// ---------------------------------------------------------------------------
__global__ __launch_bounds__(256)
void ln_score_wmma(const float* __restrict__ x, const float* __restrict__ gw,
                   const float* __restrict__ S, float* __restrict__ scores,
                   int D) {
    __shared__ float sgw[1024];
    for (int i = threadIdx.x; i < D; i += blockDim.x) sgw[i] = gw[i];
    __syncthreads();

    const int lane = threadIdx.x & 31;
    const int wave = threadIdx.x >> 5;
    const long long tbase =
        ((long long)blockIdx.x * (blockDim.x >> 5) + wave) * 16;
    const int m   = lane & 15;           // token row within tile (M)
    const int kap = (lane & 16) ? 2 : 0; // K offset of this lane-half

    const float* rowp = x + (tbase + m) * (long long)D + kap;

    v8f c = {0.f, 0.f, 0.f, 0.f, 0.f, 0.f, 0.f, 0.f};
    float s = 0.f, ss = 0.f;

    #pragma unroll 4
    for (int k0 = 0; k0 < D; k0 += 4) {
        v2f a = *(const v2f*)(rowp + k0);                 // global_load_b64
        v2f g = *(const v2f*)(&sgw[k0 + kap]);            // ds_load_b64
        s  += a.x + a.y;
        ss += a.x * a.x + a.y * a.y;
        // (neg_a, A, neg_b, B, c_mod, C, reuse_a, reuse_b)
        c = __builtin_amdgcn_wmma_f32_16x16x4_f32(
                false, a, false, g, (short)0, c, false, false);
    }

    // Combine the two K-halves of each token's sum / sumsq.
    float st  = s  + __shfl_xor(s, 16);
    float sst = ss + __shfl_xor(ss, 16);
    const float invD = 1.0f / (float)D;
    float mu  = st * invD;
    float var = sst * invD - mu * mu;
    float rs  = rsqrtf(var + LN_EPS);

    const float S1  = S[0];
    const float S2b = S[1];

    // D-matrix: VGPR v holds M=v (lanes 0-15) / M=v+8 (lanes 16-31); every
    // column is identical (B broadcast over N). Lanes 0 and 16 write 8 each.
    const int half8 = (lane & 16) ? 8 : 0;
    #pragma unroll
    for (int v = 0; v < 8; ++v) {
        float muv = __shfl(mu, v + half8);   // stats for token (v+half8)
        float rsv = __shfl(rs, v + half8);
        float sc  = rsv * (c[v] - muv * S1) + S2b;
        if ((lane & 15) == 0) scores[tbase + v + half8] = sc;
    }
}

// ---------------------------------------------------------------------------
// Kernel 2: per-batch bitonic sort of 8192 (score,idx) keys in LDS.
// key = (~monotone(float_bits) << 32) | idx  -> ascending u64 sort gives
// descending score, ascending index on ties (jax.lax.top_k order).
// ---------------------------------------------------------------------------
__global__ __launch_bounds__(1024)
void topk_sort_kernel(const float* __restrict__ scores, int* __restrict__ idx_out,
                      int N, int k) {
    __shared__ unsigned long long keys[8192];
    const int b = blockIdx.x;
    const float* s = scores + (size_t)b * N;

    for (int i = threadIdx.x; i < N; i += blockDim.x) {
        unsigned u = __float_as_uint(s[i]);
        u = (u & 0x80000000u) ? ~u : (u | 0x80000000u); // ascending map
        unsigned du = ~u;                               // descending score
        keys[i] = ((unsigned long long)du << 32) | (unsigned)i;
    }
    __syncthreads();

    for (int size = 2; size <= N; size <<= 1) {
        for (int stride = size >> 1; stride > 0; stride >>= 1) {
            for (int i = threadIdx.x; i < N; i += blockDim.x) {
                int j = i ^ stride;
                if (j > i) {
                    unsigned long long a = keys[i];
                    unsigned long long c = keys[j];
                    bool asc = ((i & size) == 0);
                    if (asc ? (a > c) : (a < c)) {
                        keys[i] = c;
                        keys[j] = a;
                    }
                }
            }
            __syncthreads();
        }
    }

    for (int i = threadIdx.x; i < k; i += blockDim.x)
        idx_out[(size_t)b * N + i] = (int)(keys[i] & 0xFFFFFFFFu);
}

// ---------------------------------------------------------------------------
// Kernel 3: gather kept rows via the CDNA5 async global<->LDS pipeline.
// One wave per output row (3 KB). Per lane: 6x GLOBAL_LOAD_ASYNC_TO_LDS_B128
// (IOFFSET applies to both LDS and global address), S_WAIT_ASYNCCNT 0, then
// 6x GLOBAL_STORE_ASYNC_FROM_LDS_B128. Data never touches VGPRs; the wave
// retires on S_ENDPGM's implicit wait-idle.
// ---------------------------------------------------------------------------
__global__ __launch_bounds__(256)
void gather_async_kernel(const float* __restrict__ x, const int* __restrict__ idx,
                         float* __restrict__ out, int D, int N, int k,
                         long long totalRows) {
    __shared__ alignas(16) char stage[8][3072];
    const int lane = threadIdx.x & 31;
    const int wave = threadIdx.x >> 5;
    long long r = (long long)blockIdx.x * (blockDim.x >> 5) + wave;
    if (r >= totalRows) return;
    int b = (int)(r / k);
    int j = (int)(r % k);
    int t = idx[(size_t)b * N + j];

    const float* srcRow = x + ((size_t)b * N + t) * (size_t)D;
    float*       dstRow = out + ((size_t)b * k + j) * (size_t)D;

    if (D == 768) {
        // low 32 bits of a flat shared pointer == workgroup-relative LDS offset
        unsigned  loff = (unsigned)(uintptr_t)(&stage[wave][lane * 16]);
        long long gsrc = (long long)(uintptr_t)((const char*)srcRow + lane * 16);
        long long gdst = (long long)(uintptr_t)((char*)dstRow + lane * 16);

        asm volatile(
            "global_load_async_to_lds_b128 %0, %1, off\n\t"
            "global_load_async_to_lds_b128 %0, %1, off offset:512\n\t"
            "global_load_async_to_lds_b128 %0, %1, off offset:1024\n\t"
            "global_load_async_to_lds_b128 %0, %1, off offset:1536\n\t"
            "global_load_async_to_lds_b128 %0, %1, off offset:2048\n\t"
            "global_load_async_to_lds_b128 %0, %1, off offset:2560\n\t"
            "s_wait_asynccnt 0x0"
            :: "v"(loff), "v"(gsrc) : "memory");

        asm volatile(
            "global_store_async_from_lds_b128 %1, %0, off\n\t"
            "global_store_async_from_lds_b128 %1, %0, off offset:512\n\t"
            "global_store_async_from_lds_b128 %1, %0, off offset:1024\n\t"
            "global_store_async_from_lds_b128 %1, %0, off offset:1536\n\t"
            "global_store_async_from_lds_b128 %1, %0, off offset:2048\n\t"
            "global_store_async_from_lds_b128 %1, %0, off offset:2560"
            :: "v"(loff), "v"(gdst) : "memory");
        // async stores drain before S_ENDPGM retires the wave (implicit wait-idle)
    } else {
        const float4* src = (const float4*)srcRow;
        float4*       dst = (float4*)dstRow;
        const int nvec = D >> 2;
        for (int it = lane; it < nvec; it += 32) dst[it] = src[it];
    }
}

// ---------------------------------------------------------------------------
extern "C" void kernel_launch(void* const* d_in, const int* in_sizes, int n_in,
                              void* d_out, int out_size, void* d_ws, size_t ws_size,
                              hipStream_t stream) {
    const float* x     = (const float*)d_in[0];
    const float* gamma = (const float*)d_in[1];
    const float* beta  = (const float*)d_in[2];
    const float* w     = (const float*)d_in[3];
    const float* bsc   = (const float*)d_in[4];

    const int D = in_sizes[1];                       // 768
    const long long BN = (long long)in_sizes[0] / D; // B*N = 131072
    const int B = 16;
    const int N = (int)(BN / B);                     // 8192
    const int k = (N * 7) / 10;                      // 5734 == int(N*0.7)

    // Workspace layout: [gw: D floats][S: 2 floats @ gw+D][pad to 1024]
    //                   [scores: B*N floats][idx: B*N ints]
    float* gw     = (float*)d_ws;
    float* S      = gw + D;
    float* scores = gw + 1024;
    int*   idxbuf = (int*)(scores + BN);

    prep_kernel<<<1, 256, 0, stream>>>(gamma, beta, w, bsc, gw, S, D);

    const int wavesPerBlock = 8;
    const int blocks1 = (int)((BN / 16) / wavesPerBlock); // 1024
    ln_score_wmma<<<blocks1, 256, 0, stream>>>(x, gw, S, scores, D);

    topk_sort_kernel<<<B, 1024, 0, stream>>>(scores, idxbuf, N, k);

    const long long rows = (long long)B * k;
    const int blocks3 = (int)((rows + wavesPerBlock - 1) / wavesPerBlock);
    gather_async_kernel<<<blocks3, 256, 0, stream>>>(x, idxbuf, (float*)d_out,
                                                     D, N, k, rows);
}